// GAT_29841432773037
// MI455X (gfx1250) — compile-verified
//
#include <hip/hip_runtime.h>
#include <math.h>

#define NN 50000
#define EE 800000
#define EPT 850000   // EE + NN self loops
#define DIN 128
#define DH 64
#define DOUT 40
#define NEG_SLOPE 0.2f

typedef __attribute__((ext_vector_type(2))) float v2f;
typedef __attribute__((ext_vector_type(8))) float v8f;

__device__ __forceinline__ float lrelu(float x) {
  return x > 0.0f ? x : NEG_SLOPE * x;
}

// order-preserving float -> uint mapping for atomicMax on floats
__device__ __forceinline__ unsigned flipf(float f) {
  unsigned u = __float_as_uint(f);
  return (u & 0x80000000u) ? ~u : (u | 0x80000000u);
}
__device__ __forceinline__ float unflipf(unsigned u) {
  unsigned b = (u & 0x80000000u) ? (u ^ 0x80000000u) : ~u;
  return __uint_as_float(b);
}

__device__ __forceinline__ void edge_nodes(const int* __restrict__ ei, int eid,
                                           int& s, int& d) {
  if (eid < EE) { s = ei[eid]; d = ei[EE + eid]; }
  else          { s = eid - EE; d = s; }
}

// ---------------------------------------------------------------------------
// GEMM via V_WMMA_F32_16X16X4_F32.  X:[NN,K] row-major, W:[K,NC] row-major,
// H:[NN,NC].  One wave computes a 16 x (NTILE*16) tile; W (zero-padded to
// NTILE*16 cols) is staged in LDS.  256 threads = 8 waves = 128 rows / block.
// ---------------------------------------------------------------------------
template <int K, int NTILE, int NC>
__global__ __launch_bounds__(256) void gemm_wmma_k(
    const float* __restrict__ X, const float* __restrict__ W,
    float* __restrict__ H) {
  constexpr int NP = NTILE * 16;
  __shared__ float Wl[K * NP];

  const int tid = threadIdx.x;
  for (int i = tid; i < K * NP; i += 256) {
    int r = i / NP, c = i % NP;
    Wl[i] = (c < NC) ? W[r * NC + c] : 0.0f;
  }
  __syncthreads();

  const int wave = tid >> 5;
  const int lane = tid & 31;
  const int half = lane >> 4;  // 0: lanes 0-15, 1: lanes 16-31
  const int lm = lane & 15;

  int m0 = (blockIdx.x * 8 + wave) * 16;
  if (m0 + 16 > NN) m0 = NN - 16;  // keep EXEC all-1s for WMMA; dup writes OK

  v8f acc[NTILE];
#pragma unroll
  for (int t = 0; t < NTILE; ++t) acc[t] = (v8f)0.0f;

  const float* __restrict__ xrow = X + (size_t)(m0 + lm) * K;
  for (int kb = 0; kb < K; kb += 4) {
    // A 16x4 f32 layout: VGPR0 = {K=0 | K=2}, VGPR1 = {K=1 | K=3} by lane half
    const float2 av = *(const float2*)(xrow + kb + 2 * half);
    v2f A;
    A[0] = av.x;
    A[1] = av.y;
    const float* wr = &Wl[(kb + 2 * half) * NP];
#pragma unroll
    for (int t = 0; t < NTILE; ++t) {
      v2f B;
      B[0] = wr[t * 16 + lm];       // B[K=2h  ][n]
      B[1] = wr[NP + t * 16 + lm];  // B[K=2h+1][n]
      acc[t] = __builtin_amdgcn_wmma_f32_16x16x4_f32(
          false, A, false, B, (short)0, acc[t], false, false);
    }
  }

#pragma unroll
  for (int t = 0; t < NTILE; ++t) {
    int col = t * 16 + lm;
    if (col < NC) {
#pragma unroll
      for (int v = 0; v < 8; ++v)
        H[(size_t)(m0 + v + 8 * half) * NC + col] = acc[t][v];
    }
  }
}

// a_s[n] = h[n,:] . att_src ; a_d[n] = h[n,:] . att_dst
template <int D>
__global__ __launch_bounds__(256) void attn_score_k(
    const float* __restrict__ h, const float* __restrict__ att_s,
    const float* __restrict__ att_d, float* __restrict__ oas,
    float* __restrict__ oad) {
  int n = blockIdx.x * 256 + threadIdx.x;
  if (n >= NN) return;
  const float* __restrict__ row = h + (size_t)n * D;
  float ss = 0.0f, sd = 0.0f;
#pragma unroll 8
  for (int c = 0; c < D; ++c) {
    float x = row[c];
    ss += x * att_s[c];
    sd += x * att_d[c];
  }
  oas[n] = ss;
  oad[n] = sd;
}

__global__ __launch_bounds__(256) void edge_max_k(
    const int* __restrict__ ei, const float* __restrict__ as,
    const float* __restrict__ ad, unsigned* __restrict__ mflip) {
  int e = blockIdx.x * 256 + threadIdx.x;
  if (e >= EPT) return;
  int s, d;
  edge_nodes(ei, e, s, d);
  float x = lrelu(as[s] + ad[d]);
  atomicMax(&mflip[d], flipf(x));
}

__global__ __launch_bounds__(256) void edge_sum_k(
    const int* __restrict__ ei, const float* __restrict__ as,
    const float* __restrict__ ad, const unsigned* __restrict__ mflip,
    float* __restrict__ p, float* __restrict__ ssum) {
  int e = blockIdx.x * 256 + threadIdx.x;
  if (e >= EPT) return;
  int s, d;
  edge_nodes(ei, e, s, d);
  float x = lrelu(as[s] + ad[d]);
  float pe = __expf(x - unflipf(mflip[d]));
  p[e] = pe;
  atomicAdd(&ssum[d], pe);
}

// out[dst, c] += (p[e] / (sum[dst]+eps)) * h[src, c]
template <int D>
__global__ __launch_bounds__(256) void edge_scatter_k(
    const int* __restrict__ ei, const float* __restrict__ p,
    const float* __restrict__ ssum, const float* __restrict__ h,
    float* __restrict__ out) {
  long long tid = (long long)blockIdx.x * 256 + threadIdx.x;
  if (tid >= (long long)EPT * D) return;
  int eid = (int)(tid / D);
  int c = (int)(tid % D);
  int s, d;
  edge_nodes(ei, eid, s, d);
  float alpha = p[eid] / (ssum[d] + 1e-16f);
  atomicAdd(&out[(size_t)d * D + c], alpha * h[(size_t)s * D + c]);
}

__global__ __launch_bounds__(256) void bias_relu_k(float* __restrict__ z,
                                                   const float* __restrict__ b) {
  int i = blockIdx.x * 256 + threadIdx.x;
  if (i >= NN * DH) return;
  z[i] = fmaxf(z[i] + b[i & (DH - 1)], 0.0f);
}

__global__ __launch_bounds__(256) void logsoftmax_k(
    const float* __restrict__ agg, const float* __restrict__ b,
    float* __restrict__ out) {
  int n = blockIdx.x * 256 + threadIdx.x;
  if (n >= NN) return;
  float v[DOUT];
  float mx = -1e30f;
#pragma unroll
  for (int c = 0; c < DOUT; ++c) {
    v[c] = agg[(size_t)n * DOUT + c] + b[c];
    mx = fmaxf(mx, v[c]);
  }
  float se = 0.0f;
#pragma unroll
  for (int c = 0; c < DOUT; ++c) se += __expf(v[c] - mx);
  float ls = __logf(se);
#pragma unroll
  for (int c = 0; c < DOUT; ++c)
    out[(size_t)n * DOUT + c] = v[c] - mx - ls;
}

extern "C" void kernel_launch(void* const* d_in, const int* in_sizes, int n_in,
                              void* d_out, int out_size, void* d_ws,
                              size_t ws_size, hipStream_t stream) {
  const float* x        = (const float*)d_in[0];
  const int*   ei       = (const int*)d_in[1];
  const float* W1       = (const float*)d_in[2];
  const float* att_src1 = (const float*)d_in[3];
  const float* att_dst1 = (const float*)d_in[4];
  const float* b1       = (const float*)d_in[5];
  const float* W2       = (const float*)d_in[6];
  const float* att_src2 = (const float*)d_in[7];
  const float* att_dst2 = (const float*)d_in[8];
  const float* b2       = (const float*)d_in[9];

  // workspace layout (floats)
  float* ws   = (float*)d_ws;
  float* h1   = ws;                           // NN*DH
  float* h2   = h1 + (size_t)NN * DH;         // NN*DOUT
  float* p1   = h2 + (size_t)NN * DOUT;       // EPT
  float* p2   = p1 + EPT;                     // EPT
  float* as1  = p2 + EPT;                     // NN
  float* ad1  = as1 + NN;                     // NN
  float* as2  = ad1 + NN;                     // NN
  float* ad2  = as2 + NN;                     // NN
  float* zb   = ad2 + NN;                     // ---- zeroed region ----
  float* agg1 = zb;                           // NN*DH
  float* agg2 = agg1 + (size_t)NN * DH;       // NN*DOUT
  unsigned* m1 = (unsigned*)(agg2 + (size_t)NN * DOUT);  // NN
  float* s1   = (float*)(m1 + NN);            // NN
  unsigned* m2 = (unsigned*)(s1 + NN);        // NN
  float* s2   = (float*)(m2 + NN);            // NN

  size_t zbytes = ((size_t)NN * (DH + DOUT) + 4 * (size_t)NN) * sizeof(float);
  hipMemsetAsync(zb, 0, zbytes, stream);

  const int gGemm = (NN / 16 + 7) / 8;          // 391 blocks (8 wave-tiles each)
  const int gEdge = (EPT + 255) / 256;          // 3321
  const int gNode = (NN + 255) / 256;           // 196
  const int gSc1  = (int)(((long long)EPT * DH + 255) / 256);
  const int gSc2  = (int)(((long long)EPT * DOUT + 255) / 256);
  const int gBR   = (NN * DH + 255) / 256;

  // ---- layer 1 ----
  gemm_wmma_k<DIN, 4, DH><<<gGemm, 256, 0, stream>>>(x, W1, h1);
  attn_score_k<DH><<<gNode, 256, 0, stream>>>(h1, att_src1, att_dst1, as1, ad1);
  edge_max_k<<<gEdge, 256, 0, stream>>>(ei, as1, ad1, m1);
  edge_sum_k<<<gEdge, 256, 0, stream>>>(ei, as1, ad1, m1, p1, s1);
  edge_scatter_k<DH><<<gSc1, 256, 0, stream>>>(ei, p1, s1, h1, agg1);
  bias_relu_k<<<gBR, 256, 0, stream>>>(agg1, b1);

  // ---- layer 2 ----
  gemm_wmma_k<DH, 3, DOUT><<<gGemm, 256, 0, stream>>>(agg1, W2, h2);
  attn_score_k<DOUT><<<gNode, 256, 0, stream>>>(h2, att_src2, att_dst2, as2, ad2);
  edge_max_k<<<gEdge, 256, 0, stream>>>(ei, as2, ad2, m2);
  edge_sum_k<<<gEdge, 256, 0, stream>>>(ei, as2, ad2, m2, p2, s2);
  edge_scatter_k<DOUT><<<gSc2, 256, 0, stream>>>(ei, p2, s2, h2, agg2);

  logsoftmax_k<<<gNode, 256, 0, stream>>>(agg2, b2, (float*)d_out);
}